// GCN_60687887892835
// MI455X (gfx1250) — compile-verified
//
#include <hip/hip_runtime.h>

#define D_IN  128
#define D_OUT 64

typedef unsigned int u32;
typedef __attribute__((ext_vector_type(2)))  float v2f;
typedef __attribute__((ext_vector_type(4)))  float v4f;
typedef __attribute__((ext_vector_type(8)))  float v8f;
typedef __attribute__((ext_vector_type(4)))  u32   u32x4;
typedef __attribute__((ext_vector_type(8)))  u32   u32x8;

// ---------------------------------------------------------------------------
// Issue one TDM 2D tile load: global (row-major, f32) -> LDS.
//   lds_off          : byte offset of destination inside workgroup LDS
//   gaddr            : global address of tile start (element [0,0] of tile)
//   dim0 / dim1      : tensor extent in elements (x, rows) for OOB clamping
//   stride0          : elements between consecutive rows in memory
//   tile_w / tile_h  : tile extent (elements, rows)
// Must be issued by a single wave; tracked with TENSORcnt.
// ---------------------------------------------------------------------------
__device__ __forceinline__ void tdm_load_tile_f32(u32 lds_off, const void* gaddr,
                                                  u32 dim0, u32 dim1, u32 stride0,
                                                  u32 tile_w, u32 tile_h) {
    const unsigned long long ga = (unsigned long long)gaddr;

    u32x4 g0;
    g0.x = 1u;                                          // count=1, user descriptor
    g0.y = lds_off;                                     // lds_addr (bytes)
    g0.z = (u32)(ga & 0xFFFFFFFFu);                     // global_addr[31:0]
    g0.w = (u32)((ga >> 32) & 0x01FFFFFFu)              // global_addr[56:32]
         | (2u << 30);                                  // type = 2 ("image")

    u32x8 g1;
    g1.s0 = (2u << 16);                                 // data_size = 4 bytes
    g1.s1 = (dim0 & 0xFFFFu) << 16;                     // tensor_dim0[15:0]
    g1.s2 = ((dim0 >> 16) & 0xFFFFu)                    // tensor_dim0[31:16]
          | ((dim1 & 0xFFFFu) << 16);                   // tensor_dim1[15:0]
    g1.s3 = ((dim1 >> 16) & 0xFFFFu)                    // tensor_dim1[31:16]
          | ((tile_w & 0xFFFFu) << 16);                 // tile_dim0
    g1.s4 = (tile_h & 0xFFFFu);                         // tile_dim1 (tile_dim2 = 0)
    g1.s5 = stride0;                                    // tensor_dim0_stride[31:0]
    g1.s6 = 0u;                                         // stride hi / dim1_stride lo
    g1.s7 = 0u;

    asm volatile("tensor_load_to_lds %0, %1"
                 :: "s"(g0), "s"(g1)
                 : "memory");
}

// ---------------------------------------------------------------------------
// Kernel 0: zero the segment-sum accumulation region (d_out[0 : N*64]).
// ---------------------------------------------------------------------------
__global__ void gcn_zero_kernel(float* __restrict__ out, int n4) {
    int i = blockIdx.x * blockDim.x + threadIdx.x;
    if (i < n4) {
        ((v4f*)out)[i] = (v4f){0.f, 0.f, 0.f, 0.f};
    }
}

// ---------------------------------------------------------------------------
// Kernel 1: support = node_x @ weights via V_WMMA_F32_16X16X4_F32.
// Block = 128 threads = 4 waves; block stages a 64x128 X tile (32 KB) and
// the whole 128x64 W (32 KB) into LDS via the Tensor Data Mover, then each
// wave computes one 16x64 output tile from LDS fragments.
// ---------------------------------------------------------------------------
__global__ void __launch_bounds__(128)
gcn_gemm_kernel(const float* __restrict__ node_x,
                const float* __restrict__ weights,
                float* __restrict__ support,
                int n_nodes) {
    __shared__ float sX[64 * D_IN];    // 32 KB: rows row0 .. row0+63
    __shared__ float sW[D_IN * D_OUT]; // 32 KB: full weight matrix

    const int lane   = threadIdx.x & 31;
    const int waveId = threadIdx.x >> 5;
    const int m      = lane & 15;      // row within A tile / col within B,D tile
    const int half   = lane >> 4;      // 0: lanes 0-15, 1: lanes 16-31
    const int rowB   = blockIdx.x * 64;

    // --- TDM stage: one wave issues both descriptors, waits TENSORcnt ---
    if (threadIdx.x < 32) {
        tdm_load_tile_f32((u32)(size_t)&sX[0],
                          node_x + (size_t)rowB * D_IN,
                          D_IN, (u32)n_nodes, D_IN, /*tile_w=*/D_IN, /*tile_h=*/64);
        tdm_load_tile_f32((u32)(size_t)&sW[0],
                          weights,
                          D_OUT, D_IN, D_OUT, /*tile_w=*/D_OUT, /*tile_h=*/D_IN);
        __builtin_amdgcn_s_wait_tensorcnt(0);
    }
    __syncthreads();   // publish LDS tiles to all 4 waves

    const int row0 = rowB + waveId * 16;
    if (row0 >= n_nodes) return;       // after the only barrier: safe

    const float* arow = &sX[(size_t)(waveId * 16 + m) * D_IN];
    const int    koff = half * 2;      // this lane's K pair within the K=4 slab

    v8f acc0 = {}, acc1 = {}, acc2 = {}, acc3 = {};

    #pragma unroll 4
    for (int kk = 0; kk < D_IN; kk += 4) {
        v2f a = *(const v2f*)(arow + kk + koff);
        const float* wk0 = &sW[(kk + koff) * D_OUT];
        const float* wk1 = wk0 + D_OUT;
        #pragma unroll
        for (int t = 0; t < 4; ++t) {
            const int bcol = t * 16 + m;
            v2f b;
            b.x = wk0[bcol];
            b.y = wk1[bcol];
            v8f* accp = (t == 0) ? &acc0 : (t == 1) ? &acc1 : (t == 2) ? &acc2 : &acc3;
            *accp = __builtin_amdgcn_wmma_f32_16x16x4_f32(
                false, a, false, b, (short)0, *accp, false, false);
        }
    }

    // D layout: VGPR r -> row (r + 8*half), lane&15 -> column
    #pragma unroll
    for (int t = 0; t < 4; ++t) {
        const v8f acc = (t == 0) ? acc0 : (t == 1) ? acc1 : (t == 2) ? acc2 : acc3;
        #pragma unroll
        for (int r = 0; r < 8; ++r) {
            const int row = row0 + r + half * 8;
            if (row < n_nodes) {
                support[(size_t)row * D_OUT + t * 16 + m] = acc[r];
            }
        }
    }
}

// ---------------------------------------------------------------------------
// Kernel 2: edge-parallel scatter.  One wave per edge; lane d handles
// features d and d+32 (two coalesced 128B rows, L2-resident atomics).
// ---------------------------------------------------------------------------
__global__ void gcn_spmm_kernel(const float* __restrict__ support,
                                const int* __restrict__ rows,
                                const int* __restrict__ cols,
                                const float* __restrict__ vals,
                                float* __restrict__ out,
                                int n_edges) {
    const int gid  = blockIdx.x * blockDim.x + threadIdx.x;
    const int e    = gid >> 5;
    const int lane = gid & 31;
    if (e >= n_edges) return;

    const int   r = rows[e];
    const int   c = cols[e];
    const float v = vals[e];

    const float* s = support + (size_t)c * D_OUT;
    float*       o = out     + (size_t)r * D_OUT;

    atomicAdd(o + lane,      v * s[lane]);
    atomicAdd(o + lane + 32, v * s[lane + 32]);
}

// ---------------------------------------------------------------------------
// Kernel 3: PReLU in place on d_out[0 : N*64].
// ---------------------------------------------------------------------------
__global__ void gcn_prelu_kernel(float* __restrict__ out,
                                 const float* __restrict__ alpha_p,
                                 int n4) {
    int i = blockIdx.x * blockDim.x + threadIdx.x;
    if (i >= n4) return;
    const float a = alpha_p[0];
    v4f x = ((v4f*)out)[i];
    #pragma unroll
    for (int j = 0; j < 4; ++j) {
        x[j] = (x[j] > 0.f) ? x[j] : a * x[j];
    }
    ((v4f*)out)[i] = x;
}

// ---------------------------------------------------------------------------
extern "C" void kernel_launch(void* const* d_in, const int* in_sizes, int n_in,
                              void* d_out, int out_size, void* d_ws, size_t ws_size,
                              hipStream_t stream) {
    const float* node_x  = (const float*)d_in[0];
    const int*   e_rows  = (const int*)  d_in[1];
    const int*   e_cols  = (const int*)  d_in[2];
    const float* e_vals  = (const float*)d_in[3];
    const float* weights = (const float*)d_in[4];
    const float* prelu_a = (const float*)d_in[5];

    const int n_nodes = in_sizes[0] / D_IN;
    const int n_edges = in_sizes[1];

    float* out     = (float*)d_out;                            // [N,64] out
    float* support = (float*)d_out + (size_t)n_nodes * D_OUT;  // [N,64] support

    // 0) zero the accumulation region
    {
        const int n4 = n_nodes * (D_OUT / 4);
        gcn_zero_kernel<<<(n4 + 255) / 256, 256, 0, stream>>>(out, n4);
    }
    // 1) support = X @ W  (TDM -> LDS -> WMMA; 4 waves / 64 rows per block)
    {
        const int tiles  = (n_nodes + 15) / 16;
        const int blocks = (tiles + 3) / 4;
        gcn_gemm_kernel<<<blocks, 128, 0, stream>>>(node_x, weights, support, n_nodes);
    }
    // 2) scatter: out[r] += v * support[c]  (one wave per edge)
    {
        const long long threads = (long long)n_edges * 32;
        const int blocks = (int)((threads + 255) / 256);
        gcn_spmm_kernel<<<blocks, 256, 0, stream>>>(support, e_rows, e_cols, e_vals,
                                                    out, n_edges);
    }
    // 3) PReLU in place
    {
        const int n4 = n_nodes * (D_OUT / 4);
        gcn_prelu_kernel<<<(n4 + 255) / 256, 256, 0, stream>>>(out, prelu_a, n4);
    }
}